// BilateralFilter_38766374814316
// MI455X (gfx1250) — compile-verified
//
#include <hip/hip_runtime.h>
#include <cstdint>
#include <cstddef>

typedef float v4f __attribute__((ext_vector_type(4)));

// Problem constants from the reference
constexpr int CH = 3;       // channels (x,y,z)
constexpr int ZD = 64;      // zenith
constexpr int AD = 512;     // azimuth
constexpr int NK = 14;      // 3x5 neighborhood minus center
constexpr int LDSW = AD + 8;  // 4-column zero halo each side (da in [-2,2], keep 8B align)
// 1/(2*theta^2): theta = [.015,.015,.01,.01] -> only two distinct values
constexpr float IC01 = 1.0f / (2.0f * 0.015f * 0.015f);  // classes 0,1
constexpr float IC23 = 1.0f / (2.0f * 0.01f * 0.01f);    // classes 2,3

__global__ __launch_bounds__(256)
void bilateral_squeezeseg_kernel(const float* __restrict__ x,
                                 float* __restrict__ out, int B) {
  // lds[c][r][col]: r = 4 padded z-rows (z0-1 .. z0+2), col = 4 + a (zero halos)
  __shared__ float lds[CH * 4 * LDSW];

  const int tid = threadIdx.x;          // 0..255
  const int z0  = blockIdx.x * 2;       // this block covers z0, z0+1
  const int b   = blockIdx.y;

  // ---- Stage padded input tile into LDS via gfx1250 async global->LDS DMA ----
  // Interior (cols 4..515) written ONLY by async b64 copies; halos/invalid rows
  // written ONLY by normal ds stores -> disjoint addresses, no DS/ASYNC hazard.
  #pragma unroll
  for (int c = 0; c < CH; ++c) {
    #pragma unroll
    for (int r = 0; r < 4; ++r) {
      const int zr = z0 - 1 + r;
      float* row = &lds[(c * 4 + r) * LDSW];
      if (zr >= 0 && zr < ZD) {
        const float* g = x + ((size_t)(b * CH + c) * ZD + zr) * AD;  // uniform
        // wave-relative LDS byte offset of this lane's 8-byte destination
        uint32_t ldsb = (uint32_t)(uintptr_t)(&row[4]) + (uint32_t)tid * 8u;
        uint32_t voff = (uint32_t)tid * 8u;  // 256 lanes * 8B = full 2048B row
        asm volatile("global_load_async_to_lds_b64 %0, %1, %2"
                     :: "v"(ldsb), "v"(voff), "s"(g) : "memory");
        if (tid < 8) row[(tid < 4) ? tid : (AD + tid)] = 0.0f;  // cols 0-3,516-519
      } else {
        for (int i = tid; i < LDSW; i += 256) row[i] = 0.0f;    // zero-pad row
      }
    }
  }
  asm volatile("s_wait_asynccnt 0x0" ::: "memory");
  __syncthreads();

  // ---- Each thread owns 4 consecutive azimuth pixels of one z row ----
  const int tz = tid >> 7;              // 0..1 -> which z row
  const int a0 = (tid & 127) * 4;       // 128 threads * 4 = 512
  const int z  = z0 + tz;

  // Register window: w[c][j][i], j = dz row, i covers a0-2 .. a0+5
  float w[CH][3][8];
  #pragma unroll
  for (int c = 0; c < CH; ++c) {
    #pragma unroll
    for (int j = 0; j < 3; ++j) {
      const float* rw = &lds[(c * 4 + tz + j) * LDSW + (a0 + 2)];
      #pragma unroll
      for (int i = 0; i < 8; ++i) w[c][j][i] = rw[i];
    }
  }

  const size_t ZA = (size_t)ZD * AD;
  float* ob = out + (size_t)(b * 4 * NK) * ZA + (size_t)z * AD + a0;

  int k = 0;
  #pragma unroll
  for (int dz = 0; dz < 3; ++dz) {
    #pragma unroll
    for (int da = 0; da < 5; ++da) {
      if (dz == 1 && da == 2) continue;   // center excluded (K = 14)
      float s[4];
      #pragma unroll
      for (int p = 0; p < 4; ++p) {
        // center x[z, a0+p]  vs  neighbor x[z+dz-1, a0+p+da-2]
        float d0 = w[0][1][p + 2] - w[0][dz][p + da];
        float d1 = w[1][1][p + 2] - w[1][dz][p + da];
        float d2 = w[2][1][p + 2] - w[2][dz][p + da];
        s[p] = d0 * d0 + d1 * d1 + d2 * d2;
      }
      v4f e0 = { __expf(-s[0] * IC01), __expf(-s[1] * IC01),
                 __expf(-s[2] * IC01), __expf(-s[3] * IC01) };
      v4f e1 = { __expf(-s[0] * IC23), __expf(-s[1] * IC23),
                 __expf(-s[2] * IC23), __expf(-s[3] * IC23) };
      float* base = ob + (size_t)k * ZA;
      // [B,4,14,Z,A]: class stride = NK*ZA. Streaming NT b128 stores.
      __builtin_nontemporal_store(e0, (v4f*)(base + (size_t)(0 * NK) * ZA));
      __builtin_nontemporal_store(e0, (v4f*)(base + (size_t)(1 * NK) * ZA));
      __builtin_nontemporal_store(e1, (v4f*)(base + (size_t)(2 * NK) * ZA));
      __builtin_nontemporal_store(e1, (v4f*)(base + (size_t)(3 * NK) * ZA));
      ++k;
    }
  }
}

extern "C" void kernel_launch(void* const* d_in, const int* in_sizes, int n_in,
                              void* d_out, int out_size, void* d_ws, size_t ws_size,
                              hipStream_t stream) {
  (void)n_in; (void)out_size; (void)d_ws; (void)ws_size;
  const float* x = (const float*)d_in[0];
  float* out = (float*)d_out;
  const int B = in_sizes[0] / (CH * ZD * AD);   // 32 for the reference shapes
  dim3 grid(ZD / 2, B);
  bilateral_squeezeseg_kernel<<<grid, 256, 0, stream>>>(x, out, B);
}